// RefineText_66443144069734
// MI455X (gfx1250) — compile-verified
//
#include <hip/hip_runtime.h>

typedef __attribute__((ext_vector_type(2))) float v2f;
typedef __attribute__((ext_vector_type(4))) float v4f;
typedef __attribute__((ext_vector_type(8))) float v8f;
typedef __attribute__((ext_vector_type(4))) int   v4i;

typedef __attribute__((address_space(1))) v4i* gptr_v4i;   // global
typedef __attribute__((address_space(3))) v4i* lptr_v4i;   // LDS

// Problem constants
constexpr int cB = 16, cT = 32, cC = 512, cE = 200, cH = 5, cNC = 97;
constexpr int cHC = cH * cC;                 // 2560

#if defined(__has_builtin)
#if __has_builtin(__builtin_amdgcn_global_load_async_to_lds_b128) && \
    __has_builtin(__builtin_amdgcn_s_wait_asynccnt)
#define USE_ASYNC_LDS 1
#endif
#endif

// -------- f32 WMMA helper: D = A(16x4) x B(4x16) + C ----------------------
// Lane layout (ISA 7.12.2, 32-bit operands, wave32):
//   A: row = lane&15, K = {2*(lane>>4), +1} in the two VGPRs
//   B: col = lane&15, K = {2*(lane>>4), +1}
//   C/D: elem i -> row = i + 8*(lane>>4), col = lane&15
__device__ __forceinline__ v8f wmma_f32(v2f a, v2f b, v8f c) {
  return __builtin_amdgcn_wmma_f32_16x16x4_f32(
      /*neg_a=*/false, a, /*neg_b=*/false, b,
      /*c_mod=*/(short)0, c, /*reuse_a=*/false, /*reuse_b=*/false);
}

// -------- embedding gather: emb[b,t,:] = lookup[y0[b,t]] ------------------
__global__ void embed_k(const int* __restrict__ y0,
                        const float* __restrict__ lookup,
                        float* __restrict__ emb) {
  int i = blockIdx.x * blockDim.x + threadIdx.x;
  if (i < cB * cT * cE) {
    int e = i % cE, bt = i / cE;
    emb[i] = lookup[(size_t)y0[bt] * cE + e];
  }
}

// -------- generic WMMA GEMM: Out = act(A @ Bm + bias) ---------------------
// One wave per block; each wave computes a 64x16 output slab (4 M-tiles
// sharing one B fragment -> 4x fewer strided B loads). M ragged (clamped
// loads, guarded stores); N ragged (clamped loads, guarded stores). K%4==0.
// headC>0: B is W[h,e,c] (H,E,C); a 16-wide N tile lies in one head.
__global__ __launch_bounds__(32) void gemm16(
    const float* __restrict__ A, int lda,
    const float* __restrict__ Bm, int ldb,
    const float* __restrict__ bias,
    float* __restrict__ Out, int ldc,
    int M, int N, int K, int relu, int headC) {
  int n0 = blockIdx.x * 16;
  int m0 = blockIdx.y * 64;
  int lane = threadIdx.x & 31;
  int rc = lane & 15;
  int kk = (lane >> 4) * 2;
  int hf = lane >> 4;

  const float* Bp = Bm;
  if (headC > 0) {
    int h = n0 / headC;
    Bp += (size_t)h * ((size_t)K * (size_t)ldb - (size_t)headC);
  }
  int n = n0 + rc;
  int ncl = (n < N) ? n : (N - 1);   // clamp: garbage stays in its own column

  const float* Ar0 = A + (size_t)(((m0 + 0 * 16 + rc) < M) ? (m0 + 0 * 16 + rc) : (M - 1)) * lda;
  const float* Ar1 = A + (size_t)(((m0 + 1 * 16 + rc) < M) ? (m0 + 1 * 16 + rc) : (M - 1)) * lda;
  const float* Ar2 = A + (size_t)(((m0 + 2 * 16 + rc) < M) ? (m0 + 2 * 16 + rc) : (M - 1)) * lda;
  const float* Ar3 = A + (size_t)(((m0 + 3 * 16 + rc) < M) ? (m0 + 3 * 16 + rc) : (M - 1)) * lda;

  v8f acc0 = {}, acc1 = {}, acc2 = {}, acc3 = {};
#pragma unroll 2
  for (int k = 0; k < K; k += 4) {
    v2f b;
    b.x = Bp[(size_t)(k + kk) * ldb + ncl];
    b.y = Bp[(size_t)(k + kk + 1) * ldb + ncl];
    v2f a0, a1, a2, a3;
    a0.x = Ar0[k + kk]; a0.y = Ar0[k + kk + 1];
    a1.x = Ar1[k + kk]; a1.y = Ar1[k + kk + 1];
    a2.x = Ar2[k + kk]; a2.y = Ar2[k + kk + 1];
    a3.x = Ar3[k + kk]; a3.y = Ar3[k + kk + 1];
    acc0 = wmma_f32(a0, b, acc0);
    acc1 = wmma_f32(a1, b, acc1);
    acc2 = wmma_f32(a2, b, acc2);
    acc3 = wmma_f32(a3, b, acc3);
  }

  if (n < N) {
    float bv = bias ? bias[n] : 0.0f;
    v8f accs[4] = {acc0, acc1, acc2, acc3};
#pragma unroll
    for (int mt = 0; mt < 4; ++mt) {
#pragma unroll
      for (int i = 0; i < 8; ++i) {
        int m = m0 + mt * 16 + hf * 8 + i;
        if (m < M) {
          float r = accs[mt][i] + bv;
          if (relu) r = fmaxf(r, 0.0f);
          Out[(size_t)m * ldc + n] = r;
        }
      }
    }
  }
}

// -------- fused attention per (b,h): scores -> softmax -> ctx -------------
// S[idx,t] = (t>idx) ? scale * Q[idx,h,:]·EK[b,t,h,:] : 0 ; softmax over t
// (masked entries contribute exp(0-max) to the denominator, matching the
// reference where q·k = q·0 = 0); ctx sums only t>idx.
// The EV slice (32x512 f32 = 64KB) is prefetched global->LDS with async
// loads, overlapping the score GEMM + softmax; phase 3 reads B from LDS.
__global__ __launch_bounds__(128) void attn_fused(
    const float* __restrict__ Q,    // (T, H*C)
    const float* __restrict__ EK,   // (B*T, H*C)
    const float* __restrict__ EV,   // (B*T, H*C)
    float* __restrict__ Ctx) {      // (T*B, H*C)
  __shared__ float S[cT * cT];
  __shared__ float EVL[cT * cC];    // 64 KB
  int b = blockIdx.x / cH;
  int h = blockIdx.x % cH;
  int tid = threadIdx.x;
  int w = tid >> 5;
  int lane = tid & 31;
  int rc = lane & 15, kk = (lane >> 4) * 2, hf = lane >> 4;

  const float* Qh = Q + h * cC;
  const float* EKb = EK + (size_t)b * cT * cHC + h * cC;
  const float* EVb = EV + (size_t)b * cT * cHC + h * cC;

  // Kick off EV slice prefetch into LDS (rows t=0..31, 512 floats each).
#ifdef USE_ASYNC_LDS
  for (int t = 0; t < cT; ++t) {
    const float* g = EVb + (size_t)t * cHC + tid * 4;
    float* l = &EVL[t * cC + tid * 4];
    __builtin_amdgcn_global_load_async_to_lds_b128(
        (gptr_v4i)(void*)const_cast<float*>(g),
        (lptr_v4i)(void*)l, 0, 0);
  }
#else
  for (int t = 0; t < cT; ++t) {
    v4f val = *(const v4f*)(EVb + (size_t)t * cHC + tid * 4);
    *(v4f*)(&EVL[t * cC + tid * 4]) = val;
  }
#endif

  // Phase 1: S(32x32) = Q_rows(32xC) @ EK_rowsT(Cx32); one 16x16 tile/wave.
  {
    int m0 = (w >> 1) * 16, n0 = (w & 1) * 16;
    const float* Arow = Qh + (size_t)(m0 + rc) * cHC;
    const float* Brow = EKb + (size_t)(n0 + rc) * cHC;
    v8f acc = {};
#pragma unroll 2
    for (int k = 0; k < cC; k += 4) {
      v2f a, bv;
      a.x = Arow[k + kk];  a.y = Arow[k + kk + 1];
      bv.x = Brow[k + kk]; bv.y = Brow[k + kk + 1];
      acc = wmma_f32(a, bv, acc);
    }
#pragma unroll
    for (int i = 0; i < 8; ++i)
      S[(m0 + hf * 8 + i) * cT + n0 + rc] = acc[i];
  }
  __syncthreads();

  // Phase 2: masked softmax, one thread per idx row.
  if (tid < cT) {
    int idx = tid;
    const float scale = 0.04419417382415922f;  // 1/sqrt(512)
    float vbuf[cT];
    float mx = 0.0f;  // masked entries are exactly 0
#pragma unroll
    for (int t = 0; t < cT; ++t) {
      float s = (t > idx) ? S[idx * cT + t] * scale : 0.0f;
      vbuf[t] = s;
      mx = fmaxf(mx, s);
    }
    float Z = 0.0f;
#pragma unroll
    for (int t = 0; t < cT; ++t) { vbuf[t] = __expf(vbuf[t] - mx); Z += vbuf[t]; }
    float inv = 1.0f / Z;
#pragma unroll
    for (int t = 0; t < cT; ++t)
      S[idx * cT + t] = (t > idx) ? vbuf[t] * inv : 0.0f;
  }
#ifdef USE_ASYNC_LDS
  __builtin_amdgcn_s_wait_asynccnt(0);   // my async EV rows are in LDS
#endif
  __syncthreads();                       // everyone's rows are in LDS

  // Phase 3: Ctx(32x512) = attn(32x32) @ EV_lds(32x512); 64 tiles / 4 waves.
  for (int ti = w; ti < 64; ti += 4) {
    int m0 = (ti & 1) * 16;
    int n0 = (ti >> 1) * 16;
    v8f acc = {};
#pragma unroll
    for (int k = 0; k < cT; k += 4) {
      v2f a, bv;
      a.x = S[(m0 + rc) * cT + k + kk];
      a.y = S[(m0 + rc) * cT + k + kk + 1];
      bv.x = EVL[(k + kk) * cC + n0 + rc];
      bv.y = EVL[(k + kk + 1) * cC + n0 + rc];
      acc = wmma_f32(a, bv, acc);
    }
#pragma unroll
    for (int i = 0; i < 8; ++i) {
      int idx = m0 + hf * 8 + i;
      Ctx[((size_t)idx * cB + b) * cHC + h * cC + n0 + rc] = acc[i];
    }
  }
}

// -------- build classifier inputs: X = [fi ; context ; 0.5(fi+context)] ---
__global__ void build_x(const float* __restrict__ bridge,   // (B*T, C)
                        const float* __restrict__ ctxt,     // (T*B, C)
                        float* __restrict__ X) {            // (3*T*B, C)
  int i = blockIdx.x * blockDim.x + threadIdx.x;
  if (i < 3 * cT * cB * cC) {
    int c = i % cC, r = i / cC;
    int j = r / (cT * cB), rb = r % (cT * cB);
    int idx = rb / cB, b = rb % cB;
    float fi = bridge[((size_t)b * cT + idx) * cC + c];
    float cx = ctxt[(size_t)rb * cC + c];
    X[i] = (j == 0) ? fi : (j == 1) ? cx : 0.5f * (fi + cx);
  }
}

// -------- (3, T, B, NC) -> 3 x (B, NC, T) transposed output ---------------
__global__ void writeout(const float* __restrict__ Sb, float* __restrict__ out) {
  int i = blockIdx.x * blockDim.x + threadIdx.x;
  if (i < 3 * cB * cNC * cT) {
    int idx = i % cT;
    int r1 = i / cT;
    int n = r1 % cNC;
    int r2 = r1 / cNC;
    int b = r2 % cB, j = r2 / cB;
    out[i] = Sb[(size_t)(j * cT * cB + idx * cB + b) * cNC + n];
  }
}

extern "C" void kernel_launch(void* const* d_in, const int* in_sizes, int n_in,
                              void* d_out, int out_size, void* d_ws, size_t ws_size,
                              hipStream_t stream) {
  const float* cf      = (const float*)d_in[0];
  // d_in[1] = cp : unused by the reference
  const int*   y0      = (const int*)d_in[2];
  const float* lookup  = (const float*)d_in[3];
  const float* vis_W   = (const float*)d_in[4];
  const float* vis_b   = (const float*)d_in[5];
  const float* Wq      = (const float*)d_in[6];
  const float* Wk      = (const float*)d_in[7];
  const float* Wv      = (const float*)d_in[8];
  const float* pos_emb = (const float*)d_in[9];
  const float* sem_W   = (const float*)d_in[10];
  const float* sem_b   = (const float*)d_in[11];
  const float* cls_W1  = (const float*)d_in[12];
  const float* cls_b1  = (const float*)d_in[13];
  const float* cls_W2  = (const float*)d_in[14];
  const float* cls_b2  = (const float*)d_in[15];
  float* out = (float*)d_out;

  // Workspace layout (floats)
  float* ws   = (float*)d_ws;
  float* emb  = ws;                                 // B*T*E      = 102400
  float* brg  = emb  + (size_t)cB * cT * cE;        // B*T*C      = 262144
  float* EK   = brg  + (size_t)cB * cT * cC;        // B*T*H*C    = 1310720
  float* EV   = EK   + (size_t)cB * cT * cHC;       // B*T*H*C    = 1310720
  float* Qb   = EV   + (size_t)cB * cT * cHC;       // T*H*C      = 81920
  float* Ctx  = Qb   + (size_t)cT * cHC;            // T*B*H*C    = 1310720
  float* ctxt = Ctx  + (size_t)cT * cB * cHC;       // T*B*C      = 262144
  float* X    = ctxt + (size_t)cT * cB * cC;        // 3*T*B*C    = 786432
  float* Hb   = X    + (size_t)3 * cT * cB * cC;    // 3*T*B*C    = 786432
  float* Sb   = Hb   + (size_t)3 * cT * cB * cC;    // 3*T*B*NC   = 148992

  // 1) emb = lookup[y0]
  embed_k<<<(cB * cT * cE + 255) / 256, 256, 0, stream>>>(y0, lookup, emb);

  // 2) bridge = cf @ vis_W + vis_b          (512 x 512 x 512)
  gemm16<<<dim3(cC / 16, (cB * cT) / 64), 32, 0, stream>>>(
      cf, cC, vis_W, cC, vis_b, brg, cC, cB * cT, cC, cC, 0, 0);

  // 3) EK/EV = emb @ Wk/Wv  per head        (512 x 2560 x 200)
  gemm16<<<dim3(cHC / 16, (cB * cT) / 64), 32, 0, stream>>>(
      emb, cE, Wk, cC, nullptr, EK, cHC, cB * cT, cHC, cE, 0, cC);
  gemm16<<<dim3(cHC / 16, (cB * cT) / 64), 32, 0, stream>>>(
      emb, cE, Wv, cC, nullptr, EV, cHC, cB * cT, cHC, cE, 0, cC);

  // 4) Q = pos_emb @ Wq per head            (32 x 2560 x 200, ragged M)
  gemm16<<<dim3(cHC / 16, 1), 32, 0, stream>>>(
      pos_emb, cE, Wq, cC, nullptr, Qb, cHC, cT, cHC, cE, 0, cC);

  // 5) fused attention: masked scores + softmax + ctx (async EV prefetch)
  attn_fused<<<cB * cH, 128, 0, stream>>>(Qb, EK, EV, Ctx);

  // 6) context = ctx @ sem_W + sem_b        (512 x 512 x 2560)
  gemm16<<<dim3(cC / 16, (cT * cB) / 64), 32, 0, stream>>>(
      Ctx, cHC, sem_W, cC, sem_b, ctxt, cC, cT * cB, cC, cHC, 0, 0);

  // 7) X = [fi ; context ; mix]             (1536 x 512)
  build_x<<<(3 * cT * cB * cC + 255) / 256, 256, 0, stream>>>(brg, ctxt, X);

  // 8) H = relu(X @ cls_W1 + b1)            (1536 x 512 x 512)
  gemm16<<<dim3(cC / 16, (3 * cT * cB) / 64), 32, 0, stream>>>(
      X, cC, cls_W1, cC, cls_b1, Hb, cC, 3 * cT * cB, cC, cC, 1, 0);

  // 9) S = H @ cls_W2 + b2                  (1536 x 97 x 512, ragged N)
  gemm16<<<dim3((cNC + 15) / 16, (3 * cT * cB) / 64), 32, 0, stream>>>(
      Hb, cC, cls_W2, cNC, cls_b2, Sb, cNC, 3 * cT * cB, cNC, cC, 0, 0);

  // 10) transpose to 3 x (B, NC, T)
  writeout<<<(3 * cB * cNC * cT + 255) / 256, 256, 0, stream>>>(Sb, out);
}